// ST_AttEdge_Exp_9259949490696
// MI455X (gfx1250) — compile-verified
//
#include <hip/hip_runtime.h>
#include <math.h>

// ---------------------------------------------------------------------------
// MI455X (gfx1250) implementation.
//  - Per-edge GEMM chain uses V_WMMA_F32_16X16X4_F32 (native f32 matrix op,
//    wave32, 16x16 C tile, K stepped by 4). 1 workgroup = 4 waves = 16 edges.
//  - Activations staged in LDS with padded strides (132 / 644 floats) to avoid
//    64-bank conflicts; weights read directly from global (L2-resident).
//  - Attention: per-triangle bias analytically reduced, segment softmax via
//    CAS float atomic-max + atomicAdd, scatter via global_atomic_add_f32.
// ---------------------------------------------------------------------------

typedef float v2f __attribute__((ext_vector_type(2)));
typedef float v8f __attribute__((ext_vector_type(8)));

#ifndef __has_builtin
#define __has_builtin(x) 0
#endif
#define HAVE_WMMA_F32 __has_builtin(__builtin_amdgcn_wmma_f32_16x16x4_f32)

#define WAVES 4          // 128-thread block for the edge-pipeline kernel
#define RS    132        // row stride (floats) for 128-wide LDS tiles (padded)
#define PLSZ  (16 * RS)  // one 16-row plane
#define RSCA  644        // row stride for the 640-wide scalar concat tile
#define RGATE 36         // row stride for the 32-wide gate tile

static constexpr int O_SCA  = 0;
static constexpr int O_V    = 16 * RSCA;          // input vector planes  (3x)
static constexpr int O_VH   = O_V  + 3 * PLSZ;    // vh planes            (3x)
static constexpr int O_VO   = O_VH + 3 * PLSZ;    // out-vector planes    (3x)
static constexpr int O_OUT  = O_VO + 3 * PLSZ;    // scalar out   [16][RS]
static constexpr int O_OUT2 = O_OUT + PLSZ;       // scalar out 2 [16][RS]
static constexpr int O_GATE = O_OUT2 + PLSZ;      // gate         [16][RGATE]
static constexpr int O_MISC = O_GATE + 16 * RGATE;
static constexpr int SMEM_FLOATS = O_MISC + 16 * 8;   // = 34240 floats (~134 KB)

// ---------------------------------------------------------------------------
// Tiled GEMM: out[16][N] = A[16][K] * W^T   (W row-major [N][K], global)
// D/A/B fragment layouts per CDNA5 ISA 7.12.2 (wave32).
// ---------------------------------------------------------------------------
__device__ __forceinline__ void gemm16(float* __restrict__ out, int ldo,
                                       const float* __restrict__ A, int lda,
                                       const float* __restrict__ W, int ldw,
                                       int N, int K) {
  const int lane = threadIdx.x & 31;
  const int wave = threadIdx.x >> 5;
  const int r    = lane & 15;
  const int hi   = lane >> 4;
  const int koff = hi * 2;
  const int ntiles = N >> 4;
  for (int nt = wave; nt < ntiles; nt += WAVES) {
    const int col = nt * 16 + r;
    const int mb  = hi * 8;
#if HAVE_WMMA_F32
    v8f acc = {0.f, 0.f, 0.f, 0.f, 0.f, 0.f, 0.f, 0.f};
    const float* Ar = A + r * lda;
    const float* Wr = W + (size_t)col * ldw;
    for (int k = 0; k < K; k += 4) {
      v2f a, b;
      a[0] = Ar[k + koff]; a[1] = Ar[k + koff + 1];
      b[0] = Wr[k + koff]; b[1] = Wr[k + koff + 1];
      acc = __builtin_amdgcn_wmma_f32_16x16x4_f32(false, a, false, b,
                                                  (short)0, acc, false, false);
    }
#pragma unroll
    for (int v = 0; v < 8; ++v) out[(mb + v) * ldo + col] = acc[v];
#else
    for (int v = 0; v < 8; ++v) {
      float s = 0.f;
      for (int k = 0; k < K; ++k)
        s += A[(mb + v) * lda + k] * W[(size_t)col * ldw + k];
      out[(mb + v) * ldo + col] = s;
    }
#endif
  }
}

// Same GEMM applied to 3 spatial planes (x,y,z); planes stride PLSZ, rows RS.
__device__ __forceinline__ void gemm3(float* __restrict__ outP,
                                      const float* __restrict__ inP,
                                      const float* __restrict__ W, int ldw,
                                      int N, int K) {
  const int lane = threadIdx.x & 31;
  const int wave = threadIdx.x >> 5;
  const int r    = lane & 15;
  const int hi   = lane >> 4;
  const int koff = hi * 2;
  const int ntiles = N >> 4;
  const int tasks  = 3 * ntiles;
  for (int t = wave; t < tasks; t += WAVES) {
    const int pl = t / ntiles;
    const int nt = t - pl * ntiles;
    const float* A = inP + pl * PLSZ;
    float* out = outP + pl * PLSZ;
    const int col = nt * 16 + r;
    const int mb  = hi * 8;
#if HAVE_WMMA_F32
    v8f acc = {0.f, 0.f, 0.f, 0.f, 0.f, 0.f, 0.f, 0.f};
    const float* Ar = A + r * RS;
    const float* Wr = W + (size_t)col * ldw;
    for (int k = 0; k < K; k += 4) {
      v2f a, b;
      a[0] = Ar[k + koff]; a[1] = Ar[k + koff + 1];
      b[0] = Wr[k + koff]; b[1] = Wr[k + koff + 1];
      acc = __builtin_amdgcn_wmma_f32_16x16x4_f32(false, a, false, b,
                                                  (short)0, acc, false, false);
    }
#pragma unroll
    for (int v = 0; v < 8; ++v) out[(mb + v) * RS + col] = acc[v];
#else
    for (int v = 0; v < 8; ++v) {
      float s = 0.f;
      for (int k = 0; k < K; ++k)
        s += A[(mb + v) * RS + k] * W[(size_t)col * ldw + k];
      out[(mb + v) * RS + col] = s;
    }
#endif
  }
}

// ------------------------- elementwise helpers -----------------------------
__device__ __forceinline__ void vnorm_sca(float* sm, int hs) {  // hid = 1<<hs
  const int hid = 1 << hs;
  for (int idx = threadIdx.x; idx < (16 << hs); idx += 128) {
    int r = idx >> hs, h = idx & (hid - 1);
    float x = sm[O_VH + 0 * PLSZ + r * RS + h];
    float y = sm[O_VH + 1 * PLSZ + r * RS + h];
    float z = sm[O_VH + 2 * PLSZ + r * RS + h];
    sm[O_SCA + r * RSCA + h] = sqrtf(x * x + y * y + z * z);
  }
}

__device__ __forceinline__ void gate_vec(float* sm, int vOff, const float* bg) {
  for (int idx = threadIdx.x; idx < 16 * 32; idx += 128) {
    int r = idx >> 5, c = idx & 31;
    float g = 1.f / (1.f + expf(-(sm[O_GATE + r * RGATE + c] + bg[c])));
#pragma unroll
    for (int i = 0; i < 3; ++i) sm[vOff + i * PLSZ + r * RS + c] *= g;
  }
}

__device__ __forceinline__ void leaky_out(float* sm) {  // leaky_relu(0.01)
  for (int idx = threadIdx.x; idx < 16 * 128; idx += 128) {
    int r = idx >> 7, c = idx & 127;
    float x = sm[O_OUT + r * RS + c];
    sm[O_OUT + r * RS + c] = (x >= 0.f) ? x : 0.01f * x;
  }
}

__device__ __forceinline__ void copy_out_to_sca(float* sm, int off) {
  for (int idx = threadIdx.x; idx < 16 * 128; idx += 128) {
    int r = idx >> 7, c = idx & 127;
    sm[O_SCA + r * RSCA + off + c] = sm[O_OUT + r * RS + c];
  }
}

__device__ __forceinline__ void vnleaky(float* sm, int xOff) {  // slope 0.2
  for (int idx = threadIdx.x; idx < 16 * 32; idx += 128) {
    int r = idx >> 5, c = idx & 31;
    float xv[3], dv[3], dot = 0.f, dsq = 0.f;
#pragma unroll
    for (int i = 0; i < 3; ++i) {
      xv[i] = sm[xOff + i * PLSZ + r * RS + c];
      dv[i] = sm[O_VH + i * PLSZ + r * RS + c];
      dot += xv[i] * dv[i]; dsq += dv[i] * dv[i];
    }
    if (dot < 0.f) {
      float f = 0.8f * dot / (dsq + 1e-9f);
#pragma unroll
      for (int i = 0; i < 3; ++i)
        sm[xOff + i * PLSZ + r * RS + c] = xv[i] - f * dv[i];
    }
  }
}

__device__ __forceinline__ void atomicMaxF(float* addr, float val) {
  unsigned* ua = (unsigned*)addr;
  unsigned old = *ua;
  while (__uint_as_float(old) < val) {
    unsigned assumed = old;
    old = atomicCAS(ua, assumed, __float_as_uint(val));
    if (old == assumed) break;
  }
}

// ---------------------------------------------------------------------------
struct EdgeParams {
  const float *ee3;
  const float *pWv1, *pWs, *pWg, *pbg, *pWv2, *pWd;   // nn_edge_p
  const float *lWv1, *lWs, *lWg, *lbg, *lWv2;          // nn_edge_l
  const float *eWv1, *eWs, *eWg, *ebg, *eWv2, *eWd;    // ef_p
  const float *fWv1, *fWs, *fWg, *fbg, *fWv2;          // ef_l
  const float *qWv1, *qWs, *qWg, *qbg, *qWv2;
  const float *kWv1, *kWs, *kWg, *kbg, *kWv2;
  const float *vWv1, *vWs, *vWg, *vbg, *vWv2;
};

// K0: init segment-max/-sum/accumulators and precompute bias-path vectors.
__global__ void k0_init(float* ms, float* mv, float* ss, float* sv,
                        float* accS, float* accV, float* w1e, float* w2e,
                        const float* bWv1, const float* bee, const float* bWv2,
                        int E) {
  long long tid = (long long)blockIdx.x * 256 + threadIdx.x;
  long long n1 = (long long)E * 128, n2 = n1 + (long long)E * 96;
  long long n3 = n2 + (long long)E * 4, n4 = n3 + (long long)E * 4;
  long long n5 = n4 + (long long)E * 4, n6 = n5 + (long long)E * 4;
  if (tid < n1) accS[tid] = 0.f;
  else if (tid < n2) accV[tid - n1] = 0.f;
  else if (tid < n3) ms[tid - n2] = -__builtin_huge_valf();
  else if (tid < n4) mv[tid - n3] = -__builtin_huge_valf();
  else if (tid < n5) ss[tid - n4] = 0.f;
  else if (tid < n6) sv[tid - n5] = 0.f;
  if (blockIdx.x == 0) {
    int t = threadIdx.x;
    if (t < 32) {                 // w1e[h] = <Wv1[h,:], ee>
      float a = 0.f;
      for (int c = 0; c < 32; ++c) a += bWv1[t * 32 + c] * bee[c];
      w1e[t] = a;
    } else if (t < 36) {          // w2e[o] = sum_h w1e[h]*Wv2[o,h]
      int o = t - 32; float a = 0.f;
      for (int h = 0; h < 32; ++h) {
        float w = 0.f;
        for (int c = 0; c < 32; ++c) w += bWv1[h * 32 + c] * bee[c];
        a += w * bWv2[o * 32 + h];
      }
      w2e[o] = a;
    }
  }
}

// K1: full per-edge pipeline, WMMA-tiled, 16 edges per block.
__global__ void __launch_bounds__(128)
k1_edge_pipeline(const float* __restrict__ hsca, const float* __restrict__ hvec,
                 const float* __restrict__ posq, const int* __restrict__ eidx,
                 const float* __restrict__ cpos, const float* __restrict__ nsca,
                 const float* __restrict__ nvec, EdgeParams P,
                 float* sres, float* vres, float* qs, float* qv,
                 float* ks, float* kv, float* vs, float* vv, int E) {
  extern __shared__ float sm[];
  const int tid = threadIdx.x;
  const int e0  = blockIdx.x * 16;

  // ---- phase 0: geometry, gaussian smearing (64), edge expansion ----
  if (tid < 16) {
    int e = min(e0 + tid, E - 1);
    int i = eidx[e], j = eidx[E + e];
    float vx = posq[i * 3 + 0] - cpos[j * 3 + 0];
    float vy = posq[i * 3 + 1] - cpos[j * 3 + 1];
    float vz = posq[i * 3 + 2] - cpos[j * 3 + 2];
    float L  = sqrtf(vx * vx + vy * vy + vz * vz);
    float inv = 1.f / (L + 1e-7f);
    sm[O_MISC + tid * 8 + 0] = vx * inv;
    sm[O_MISC + tid * 8 + 1] = vy * inv;
    sm[O_MISC + tid * 8 + 2] = vz * inv;
    sm[O_MISC + tid * 8 + 3] = L;
    sm[O_MISC + tid * 8 + 4] = __int_as_float(i);
    sm[O_MISC + tid * 8 + 5] = __int_as_float(j);
  }
  __syncthreads();
  {
    const float step = 3.f / 63.f;
    const float cg   = -0.5f / (step * step);
    for (int idx = tid; idx < 16 * 64; idx += 128) {
      int r = idx >> 6, k = idx & 63;
      float d = sm[O_MISC + r * 8 + 3] - step * (float)k;
      sm[O_SCA + r * RSCA + 64 + k] = expf(cg * d * d);
      float ee = P.ee3[k];
#pragma unroll
      for (int i2 = 0; i2 < 3; ++i2)
        sm[O_V + i2 * PLSZ + r * RS + k] = sm[O_MISC + r * 8 + i2] * ee;
    }
  }
  __syncthreads();

  // ======== stage A: nn_edge_p (perceptron)  64/64 -> 128/32, hid 64 =======
  gemm3(sm + O_VH, sm + O_V, P.pWv1, 64, 64, 64);
  __syncthreads();
  vnorm_sca(sm, 6);
  __syncthreads();
  gemm16(sm + O_OUT, RS, sm + O_SCA, RSCA, P.pWs, 128, 128, 128);
  __syncthreads();
  gemm16(sm + O_GATE, RGATE, sm + O_OUT, RS, P.pWg, 128, 32, 128);
  gemm3(sm + O_VO, sm + O_VH, P.pWv2, 64, 32, 64);
  __syncthreads();
  gate_vec(sm, O_VO, P.pbg);
  leaky_out(sm);
  __syncthreads();
  gemm3(sm + O_VH, sm + O_VO, P.pWd, 32, 32, 32);
  __syncthreads();
  vnleaky(sm, O_VO);
  __syncthreads();

  // ======== stage B: nn_edge_l  128/32 -> 128/32, hid 32 ==================
  copy_out_to_sca(sm, 32);
  gemm3(sm + O_VH, sm + O_VO, P.lWv1, 32, 32, 32);
  __syncthreads();
  vnorm_sca(sm, 5);
  __syncthreads();
  gemm16(sm + O_OUT, RS, sm + O_SCA, RSCA, P.lWs, 160, 128, 160);
  __syncthreads();
  gemm16(sm + O_GATE, RGATE, sm + O_OUT, RS, P.lWg, 128, 32, 128);
  gemm3(sm + O_V, sm + O_VH, P.lWv2, 32, 32, 32);
  __syncthreads();
  gate_vec(sm, O_V, P.lbg);
  __syncthreads();

  // ======== concat for ef_p ===============================================
  // NOTE: reference concat is 640/160 wide but ef_p params are sized 512/128;
  // we follow the param shapes: sca=[h(256)|node[:128]|s(128)],
  // vec=[hvec(64)|nvec[:32]|v(32)].
  for (int idx = tid; idx < 16 * 256; idx += 128) {
    int r = idx >> 8, c = idx & 255;
    int ih = __float_as_int(sm[O_MISC + r * 8 + 4]);
    sm[O_SCA + r * RSCA + 128 + c] = hsca[(size_t)ih * 256 + c];
  }
  for (int idx = tid; idx < 16 * 128; idx += 128) {
    int r = idx >> 7, c = idx & 127;
    int jh = __float_as_int(sm[O_MISC + r * 8 + 5]);
    sm[O_SCA + r * RSCA + 384 + c] = nsca[(size_t)jh * 256 + c];
    sm[O_SCA + r * RSCA + 512 + c] = sm[O_OUT + r * RS + c];
  }
  for (int idx = tid; idx < 16 * 64; idx += 128) {
    int r = idx >> 6, c = idx & 63;
    int ih = __float_as_int(sm[O_MISC + r * 8 + 4]);
#pragma unroll
    for (int i = 0; i < 3; ++i)
      sm[O_VO + i * PLSZ + r * RS + c] = hvec[((size_t)ih * 64 + c) * 3 + i];
  }
  for (int idx = tid; idx < 16 * 32; idx += 128) {
    int r = idx >> 5, c = idx & 31;
    int jh = __float_as_int(sm[O_MISC + r * 8 + 5]);
#pragma unroll
    for (int i = 0; i < 3; ++i) {
      sm[O_VO + i * PLSZ + r * RS + 64 + c] = nvec[((size_t)jh * 64 + c) * 3 + i];
      sm[O_VO + i * PLSZ + r * RS + 96 + c] = sm[O_V + i * PLSZ + r * RS + c];
    }
  }
  __syncthreads();

  // ======== stage C: ef_p (perceptron)  512/128 -> 128/32, hid 128 ========
  gemm3(sm + O_VH, sm + O_VO, P.eWv1, 128, 128, 128);
  __syncthreads();
  vnorm_sca(sm, 7);
  __syncthreads();
  gemm16(sm + O_OUT, RS, sm + O_SCA, RSCA, P.eWs, 640, 128, 640);
  __syncthreads();
  gemm16(sm + O_GATE, RGATE, sm + O_OUT, RS, P.eWg, 128, 32, 128);
  gemm3(sm + O_V, sm + O_VH, P.eWv2, 128, 32, 128);
  __syncthreads();
  gate_vec(sm, O_V, P.ebg);
  leaky_out(sm);
  __syncthreads();
  gemm3(sm + O_VH, sm + O_V, P.eWd, 32, 32, 32);
  __syncthreads();
  vnleaky(sm, O_V);
  __syncthreads();

  // ======== stage D: ef_l  128/32 -> 128/32, hid 32 =======================
  copy_out_to_sca(sm, 32);
  gemm3(sm + O_VH, sm + O_V, P.fWv1, 32, 32, 32);
  __syncthreads();
  vnorm_sca(sm, 5);
  __syncthreads();
  gemm16(sm + O_OUT, RS, sm + O_SCA, RSCA, P.fWs, 160, 128, 160);
  __syncthreads();
  gemm16(sm + O_GATE, RGATE, sm + O_OUT, RS, P.fWg, 128, 32, 128);
  gemm3(sm + O_VO, sm + O_VH, P.fWv2, 32, 32, 32);
  __syncthreads();
  gate_vec(sm, O_VO, P.fbg);
  __syncthreads();

  // store residuals s,v  and refresh sca part for the q/k/v projections
  for (int idx = tid; idx < 16 * 128; idx += 128) {
    int r = idx >> 7, c = idx & 127;
    int e = e0 + r;
    if (e < E) sres[(size_t)e * 128 + c] = sm[O_OUT + r * RS + c];
  }
  for (int idx = tid; idx < 16 * 32; idx += 128) {
    int r = idx >> 5, c = idx & 31;
    int e = e0 + r;
    if (e < E)
#pragma unroll
      for (int i = 0; i < 3; ++i)
        vres[((size_t)e * 32 + c) * 3 + i] = sm[O_VO + i * PLSZ + r * RS + c];
  }
  copy_out_to_sca(sm, 32);
  __syncthreads();

  // ======== stage E: q/k/v projections  128/32 -> 128/32, hid 32 ==========
  const float* Wv1s[3] = {P.qWv1, P.kWv1, P.vWv1};
  const float* Wss [3] = {P.qWs,  P.kWs,  P.vWs};
  const float* Wgs [3] = {P.qWg,  P.kWg,  P.vWg};
  const float* bgs [3] = {P.qbg,  P.kbg,  P.vbg};
  const float* Wv2s[3] = {P.qWv2, P.kWv2, P.vWv2};
  float* oS[3] = {qs, ks, vs};
  float* oV[3] = {qv, kv, vv};
#pragma unroll 1
  for (int pj = 0; pj < 3; ++pj) {
    gemm3(sm + O_VH, sm + O_VO, Wv1s[pj], 32, 32, 32);
    __syncthreads();
    vnorm_sca(sm, 5);
    __syncthreads();
    gemm16(sm + O_OUT2, RS, sm + O_SCA, RSCA, Wss[pj], 160, 128, 160);
    __syncthreads();
    gemm16(sm + O_GATE, RGATE, sm + O_OUT2, RS, Wgs[pj], 128, 32, 128);
    gemm3(sm + O_V, sm + O_VH, Wv2s[pj], 32, 32, 32);
    __syncthreads();
    gate_vec(sm, O_V, bgs[pj]);
    __syncthreads();
    for (int idx = tid; idx < 16 * 128; idx += 128) {
      int r = idx >> 7, c = idx & 127;
      int e = e0 + r;
      if (e < E) oS[pj][(size_t)e * 128 + c] = sm[O_OUT2 + r * RS + c];
    }
    for (int idx = tid; idx < 16 * 32; idx += 128) {
      int r = idx >> 5, c = idx & 31;
      int e = e0 + r;
      if (e < E)
#pragma unroll
        for (int i = 0; i < 3; ++i)
          oV[pj][((size_t)e * 32 + c) * 3 + i] = sm[O_V + i * PLSZ + r * RS + c];
    }
    __syncthreads();
  }
}

// K2: per-triangle attention logits (+ bias) and segment max.
__global__ void k2_alpha(const int* __restrict__ atten, const int* __restrict__ tri,
                         const float* __restrict__ tfeat, const float* __restrict__ cpos,
                         const float* __restrict__ qs, const float* __restrict__ ksp,
                         const float* __restrict__ qv, const float* __restrict__ kvp,
                         const float* __restrict__ Wsb, const float* __restrict__ Wgb,
                         const float* __restrict__ bgb,
                         const float* __restrict__ w1e, const float* __restrict__ w2e,
                         float* asr, float* avr, float* ms, float* mv,
                         int Nt, int E) {
  int t = blockIdx.x * 256 + threadIdx.x;
  if (t >= Nt) return;
  int a = tri[t], b = tri[Nt + t];
  float tx = cpos[a * 3 + 0] - cpos[b * 3 + 0];
  float ty = cpos[a * 3 + 1] - cpos[b * 3 + 1];
  float tz = cpos[a * 3 + 2] - cpos[b * 3 + 2];
  float L = sqrtf(tx * tx + ty * ty + tz * tz);
  float nv = L / (L + 1e-7f);
  float acc[4] = {0.f, 0.f, 0.f, 0.f};
  for (int h = 0; h < 32; ++h) {             // vnorm part of Ws
    float vn = nv * fabsf(w1e[h]);
#pragma unroll
    for (int o = 0; o < 4; ++o) acc[o] += Wsb[o * 160 + h] * vn;
  }
  const float step = 10.f / 123.f;
  const float cg   = -0.5f / (step * step);
  for (int k = 0; k < 124; ++k) {            // gaussian smearing part
    float d = L - step * (float)k;
    float g = expf(cg * d * d);
#pragma unroll
    for (int o = 0; o < 4; ++o) acc[o] += Wsb[o * 160 + 32 + k] * g;
  }
#pragma unroll
  for (int k = 0; k < 4; ++k) {              // tri_edge_feat part
    float f = tfeat[(size_t)t * 4 + k];
#pragma unroll
    for (int o = 0; o < 4; ++o) acc[o] += Wsb[o * 160 + 156 + k] * f;
  }
  float bv[4];
#pragma unroll
  for (int o = 0; o < 4; ++o) {              // gate + |gated out_vec|^2
    float z = bgb[o];
#pragma unroll
    for (int o2 = 0; o2 < 4; ++o2) z += acc[o2] * Wgb[o * 4 + o2];
    float g = 1.f / (1.f + expf(-z));
    float q = g * w2e[o] * nv;
    bv[o] = q * q;
  }
  int ei = atten[t], ej = atten[Nt + t];
#pragma unroll
  for (int h = 0; h < 4; ++h) {
    float s = acc[h];
    const float* qp = qs  + (size_t)ei * 128 + h * 32;
    const float* kp = ksp + (size_t)ej * 128 + h * 32;
    for (int d = 0; d < 32; ++d) s += qp[d] * kp[d];
    float v = bv[h];
    const float* qvp = qv  + (size_t)ei * 96 + h * 24;
    const float* kvq = kvp + (size_t)ej * 96 + h * 24;
    for (int d = 0; d < 24; ++d) v += qvp[d] * kvq[d];
    asr[(size_t)t * 4 + h] = s;
    avr[(size_t)t * 4 + h] = v;
    atomicMaxF(&ms[(size_t)ei * 4 + h], s);
    atomicMaxF(&mv[(size_t)ei * 4 + h], v);
  }
}

// K2b: exp(x - max) and segment sums.
__global__ void k2b_expsum(const int* __restrict__ atten, float* asr, float* avr,
                           const float* __restrict__ ms, const float* __restrict__ mv,
                           float* ss, float* sv, long long Nt4) {
  long long id = (long long)blockIdx.x * 256 + threadIdx.x;
  if (id >= Nt4) return;
  int t = (int)(id >> 2), h = (int)(id & 3);
  int ei = atten[t];
  float es = expf(asr[id] - ms[(size_t)ei * 4 + h]);
  asr[id] = es; atomicAdd(&ss[(size_t)ei * 4 + h], es);
  float ev = expf(avr[id] - mv[(size_t)ei * 4 + h]);
  avr[id] = ev; atomicAdd(&sv[(size_t)ei * 4 + h], ev);
}

// K3: softmax-weighted scatter-add (one wave per triangle).
__global__ void k3_scatter(const int* __restrict__ atten,
                           const float* __restrict__ asr, const float* __restrict__ avr,
                           const float* __restrict__ ss, const float* __restrict__ sv,
                           const float* __restrict__ vsp, const float* __restrict__ vvp,
                           float* accS, float* accV, int Nt) {
  int wave = threadIdx.x >> 5, lane = threadIdx.x & 31;
  int t = blockIdx.x * 8 + wave;
  if (t >= Nt) return;
  int ei = atten[t], ej = atten[Nt + t];
  for (int u = lane; u < 224; u += 32) {
    if (u < 128) {
      int h = u >> 5;
      float w = asr[(size_t)t * 4 + h] / (ss[(size_t)ei * 4 + h] + 1e-16f);
      atomicAdd(&accS[(size_t)ei * 128 + u], w * vsp[(size_t)ej * 128 + u]);
    } else {
      int u2 = u - 128;
      int h = u2 / 24;
      float w = avr[(size_t)t * 4 + h] / (sv[(size_t)ei * 4 + h] + 1e-16f);
      atomicAdd(&accV[(size_t)ei * 96 + u2], w * vvp[(size_t)ej * 96 + u2]);
    }
  }
}

// K4: residual + layernorms + pred head (one wave per edge).
__global__ void k4_final(const float* __restrict__ sres, const float* __restrict__ vres,
                         const float* __restrict__ accS, const float* __restrict__ accV,
                         const float* __restrict__ lng, const float* __restrict__ lnb,
                         const float* __restrict__ lvg, const float* __restrict__ lvb,
                         const float* __restrict__ predWs, const float* __restrict__ predWv1,
                         float* out, int E) {
  int wave = threadIdx.x >> 5, lane = threadIdx.x & 31;
  int e = blockIdx.x * 8 + wave;
  if (e >= E) return;
  // ---- scalar layernorm over 128 ----
  float xs[4]; float sum = 0.f;
#pragma unroll
  for (int j = 0; j < 4; ++j) {
    size_t o = (size_t)e * 128 + lane * 4 + j;
    xs[j] = sres[o] + accS[o]; sum += xs[j];
  }
  for (int m = 16; m; m >>= 1) sum += __shfl_xor(sum, m);
  float mean = sum / 128.f, vsum = 0.f;
#pragma unroll
  for (int j = 0; j < 4; ++j) { float d = xs[j] - mean; vsum += d * d; }
  for (int m = 16; m; m >>= 1) vsum += __shfl_xor(vsum, m);
  float inv = rsqrtf(vsum / 128.f + 1e-5f);
  float sn[4];
#pragma unroll
  for (int j = 0; j < 4; ++j)
    sn[j] = (xs[j] - mean) * inv * lng[lane * 4 + j] + lnb[lane * 4 + j];
  // ---- vector layernorm over (32,3) ----
  float xv[3]; float sum2 = 0.f;
#pragma unroll
  for (int i = 0; i < 3; ++i) {
    size_t o = (size_t)e * 96 + lane * 3 + i;
    xv[i] = vres[o] + accV[o]; sum2 += xv[i];
  }
  for (int m = 16; m; m >>= 1) sum2 += __shfl_xor(sum2, m);
  float mean2 = sum2 / 96.f, vsum2 = 0.f;
#pragma unroll
  for (int i = 0; i < 3; ++i) { float d = xv[i] - mean2; vsum2 += d * d; }
  for (int m = 16; m; m >>= 1) vsum2 += __shfl_xor(vsum2, m);
  float inv2 = rsqrtf(vsum2 / 96.f + 1e-5f);
  float vn3[3];
#pragma unroll
  for (int i = 0; i < 3; ++i)
    vn3[i] = (xv[i] - mean2) * inv2 * lvg[lane * 3 + i] + lvb[lane * 3 + i];
  // ---- pred gb_linear (only out_sca needed): vh = v @ Wv1^T, |vh|, Ws ----
  float vh[3] = {0.f, 0.f, 0.f};
  for (int c = 0; c < 32; ++c) {
    float w = predWv1[lane * 32 + c];
#pragma unroll
    for (int i = 0; i < 3; ++i) vh[i] += w * __shfl(vn3[i], c);
  }
  float vhn = sqrtf(vh[0] * vh[0] + vh[1] * vh[1] + vh[2] * vh[2]);
  float part[8];
#pragma unroll
  for (int o = 0; o < 8; ++o) {
    float a = predWs[o * 160 + lane] * vhn;
#pragma unroll
    for (int j = 0; j < 4; ++j)
      a += predWs[o * 160 + 32 + lane * 4 + j] * sn[j];
    for (int m = 16; m; m >>= 1) a += __shfl_xor(a, m);
    part[o] = a;
  }
  if (lane < 4)       out[(size_t)e * 4 + lane] = part[lane];
  else if (lane < 8)  out[(size_t)E * 4 + (size_t)e * 4 + (lane - 4)] = part[lane];
}

// ---------------------------------------------------------------------------
extern "C" void kernel_launch(void* const* d_in, const int* in_sizes, int n_in,
                              void* d_out, int out_size, void* d_ws, size_t ws_size,
                              hipStream_t stream) {
  const float* hsca  = (const float*)d_in[0];
  const float* hvec  = (const float*)d_in[1];
  const float* posq  = (const float*)d_in[2];
  const int*   eidx  = (const int*)d_in[3];
  const float* cpos  = (const float*)d_in[4];
  const float* nsca  = (const float*)d_in[5];
  const float* nvec  = (const float*)d_in[6];
  const int*   atten = (const int*)d_in[7];
  const int*   tri   = (const int*)d_in[8];
  const float* tfeat = (const float*)d_in[9];

  // params pytree leaves (sorted-key order).
  const float* par[53];
  for (int i = 0; i < 53; ++i) par[i] = (10 + i < n_in) ? (const float*)d_in[10 + i] : nullptr;
  const float *bias_ee = par[0];
  const float *bgl_Wg = par[1], *bgl_Ws = par[2], *bgl_Wv1 = par[3], *bgl_Wv2 = par[4], *bgl_bg = par[5];
  const float *k_Wg = par[6], *k_Ws = par[7], *k_Wv1 = par[8], *k_Wv2 = par[9], *k_bg = par[10];
  const float *ln_sca_b = par[11], *ln_sca_g = par[12], *ln_vec_b = par[13], *ln_vec_g = par[14];
  const float *q_Wg = par[15], *q_Ws = par[16], *q_Wv1 = par[17], *q_Wv2 = par[18], *q_bg = par[19];
  const float *v_Wg = par[20], *v_Ws = par[21], *v_Wv1 = par[22], *v_Wv2 = par[23], *v_bg = par[24];
  const float *ee3 = par[25];
  const float *efl_Wg = par[26], *efl_Ws = par[27], *efl_Wv1 = par[28], *efl_Wv2 = par[29], *efl_bg = par[30];
  const float *efp_Wd = par[31], *efp_Wg = par[32], *efp_Ws = par[33], *efp_Wv1 = par[34], *efp_Wv2 = par[35], *efp_bg = par[36];
  const float *nel_Wg = par[37], *nel_Ws = par[38], *nel_Wv1 = par[39], *nel_Wv2 = par[40], *nel_bg = par[41];
  const float *nep_Wd = par[42], *nep_Wg = par[43], *nep_Ws = par[44], *nep_Wv1 = par[45], *nep_Wv2 = par[46], *nep_bg = par[47];
  const float *pred_Ws = par[49], *pred_Wv1 = par[50];
  (void)ws_size; (void)out_size;

  const int E  = in_sizes[3] / 2;
  const int Nt = in_sizes[7] / 2;

  // workspace layout (floats)
  float* w = (float*)d_ws;
  size_t off = 0;
  float* sres = w + off; off += (size_t)E * 128;
  float* vres = w + off; off += (size_t)E * 96;
  float* qs   = w + off; off += (size_t)E * 128;
  float* qv   = w + off; off += (size_t)E * 96;
  float* ksb  = w + off; off += (size_t)E * 128;
  float* kvb  = w + off; off += (size_t)E * 96;
  float* vsb  = w + off; off += (size_t)E * 128;
  float* vvb  = w + off; off += (size_t)E * 96;
  float* asr  = w + off; off += (size_t)Nt * 4;
  float* avr  = w + off; off += (size_t)Nt * 4;
  float* ms   = w + off; off += (size_t)E * 4;
  float* mv   = w + off; off += (size_t)E * 4;
  float* ss   = w + off; off += (size_t)E * 4;
  float* sv   = w + off; off += (size_t)E * 4;
  float* accS = w + off; off += (size_t)E * 128;
  float* accV = w + off; off += (size_t)E * 96;
  float* w1e  = w + off; off += 32;
  float* w2e  = w + off; off += 4;

  EdgeParams P;
  P.ee3 = ee3;
  P.pWv1 = nep_Wv1; P.pWs = nep_Ws; P.pWg = nep_Wg; P.pbg = nep_bg; P.pWv2 = nep_Wv2; P.pWd = nep_Wd;
  P.lWv1 = nel_Wv1; P.lWs = nel_Ws; P.lWg = nel_Wg; P.lbg = nel_bg; P.lWv2 = nel_Wv2;
  P.eWv1 = efp_Wv1; P.eWs = efp_Ws; P.eWg = efp_Wg; P.ebg = efp_bg; P.eWv2 = efp_Wv2; P.eWd = efp_Wd;
  P.fWv1 = efl_Wv1; P.fWs = efl_Ws; P.fWg = efl_Wg; P.fbg = efl_bg; P.fWv2 = efl_Wv2;
  P.qWv1 = q_Wv1; P.qWs = q_Ws; P.qWg = q_Wg; P.qbg = q_bg; P.qWv2 = q_Wv2;
  P.kWv1 = k_Wv1; P.kWs = k_Ws; P.kWg = k_Wg; P.kbg = k_bg; P.kWv2 = k_Wv2;
  P.vWv1 = v_Wv1; P.vWs = v_Ws; P.vWg = v_Wg; P.vbg = v_bg; P.vWv2 = v_Wv2;

  const int smemBytes = SMEM_FLOATS * (int)sizeof(float);
  (void)hipFuncSetAttribute((const void*)k1_edge_pipeline,
                            hipFuncAttributeMaxDynamicSharedMemorySize, smemBytes);

  { // K0
    long long n = (long long)E * 240;
    int grid = (int)((n + 255) / 256);
    if (grid < 1) grid = 1;
    k0_init<<<grid, 256, 0, stream>>>(ms, mv, ss, sv, accS, accV, w1e, w2e,
                                      bgl_Wv1, bias_ee, bgl_Wv2, E);
  }
  { // K1
    int grid = (E + 15) / 16;
    k1_edge_pipeline<<<grid, 128, smemBytes, stream>>>(
        hsca, hvec, posq, eidx, cpos, nsca, nvec, P,
        sres, vres, qs, qv, ksb, kvb, vsb, vvb, E);
  }
  { // K2
    int grid = (Nt + 255) / 256;
    k2_alpha<<<grid, 256, 0, stream>>>(atten, tri, tfeat, cpos, qs, ksb, qv, kvb,
                                       bgl_Ws, bgl_Wg, bgl_bg, w1e, w2e,
                                       asr, avr, ms, mv, Nt, E);
  }
  { // K2b
    long long n4 = (long long)Nt * 4;
    int grid = (int)((n4 + 255) / 256);
    k2b_expsum<<<grid, 256, 0, stream>>>(atten, asr, avr, ms, mv, ss, sv, n4);
  }
  { // K3
    int grid = (Nt + 7) / 8;
    k3_scatter<<<grid, 256, 0, stream>>>(atten, asr, avr, ss, sv, vsb, vvb,
                                         accS, accV, Nt);
  }
  { // K4
    int grid = (E + 7) / 8;
    k4_final<<<grid, 256, 0, stream>>>(sres, vres, accS, accV,
                                       ln_sca_g, ln_sca_b, ln_vec_g, ln_vec_b,
                                       pred_Ws, pred_Wv1, (float*)d_out, E);
  }
}